// WaveletTransform1D_85581518340116
// MI455X (gfx1250) — compile-verified
//
#include <hip/hip_runtime.h>
#include <stdint.h>

// Wavelet transform, (32,64,8192) fp32 -> (32,64,3,8192) fp32.
// Memory-bound: ~265 MB total traffic -> ~11.4 us floor @ 23.3 TB/s.
// One row per workgroup, fully fused in LDS; input DMA'd via CDNA5 TDM
// (tensor_load_to_lds + s_wait_tensorcnt); outputs written as non-temporal
// b128 stores (write-once data, keep it out of L2).

#define L0  8192
#define NT  256

typedef unsigned int u32x4 __attribute__((ext_vector_type(4)));
typedef int          i32x4 __attribute__((ext_vector_type(4)));
typedef int          i32x8 __attribute__((ext_vector_type(8)));
typedef float        f32x4 __attribute__((ext_vector_type(4)));

#if defined(__has_builtin)
#  if __has_builtin(__builtin_amdgcn_tensor_load_to_lds)
#    define USE_TDM 1
#  endif
#endif
#ifndef USE_TDM
#  define USE_TDM 0
#endif

// zero-padded read
__device__ __forceinline__ float zget(const float* a, int n, int i) {
    return ((unsigned)i < (unsigned)n) ? a[i] : 0.0f;
}

// Evaluate 2x linear upsample (align_corners=False) of p[0..n) at output index j (0..2n)
__device__ __forceinline__ float up2_eval(const float* p, int n, int j) {
    float q = 0.5f * (float)j - 0.25f;
    q = q < 0.0f ? 0.0f : q;
    int lo = (int)q;
    int hi = lo + 1; if (hi > n - 1) hi = n - 1;
    float w = q - (float)lo;
    float a = p[lo], b = p[hi];
    return a + w * (b - a);
}

__global__ __launch_bounds__(NT)
void wavelet_fused_kernel(const float* __restrict__ x,
                          const float* __restrict__ w0,
                          const float* __restrict__ w1,
                          const float* __restrict__ w2,
                          float* __restrict__ out)
{
    __shared__ float lds[12288];          // 48 KB
    float* buf0 = lds;                    // [0,8192)    : input row, later reused
    float* s1   = lds + 8192;             // [8192,12288): avgpool level 1 (4096)
    float* f1   = buf0;                   // conv5 output (4096), reuse of buf0
    float* p1   = buf0 + 4096;            // avgpool of f1 (2048)
    float* s2   = buf0 + 6144;            // avgpool level 2 (2048)
    float* f2   = s1;                     // conv7 output (2048), reuse of s1
    float* p2   = s1 + 2048;              // avgpool of f2 (1024)

    const int r   = blockIdx.x;
    const int tid = threadIdx.x;
    const float* __restrict__ src = x + (size_t)r * L0;
    float* __restrict__ dst = out + (size_t)r * 3 * L0;

    // ---------------- Load row -> LDS via Tensor Data Mover ----------------
#if USE_TDM
    if (__builtin_amdgcn_readfirstlane((int)threadIdx.x) == 0) {
        // wave-uniform branch: only wave 0 issues the TDM op (TDM ignores EXEC)
        uint64_t ga = (uint64_t)(uintptr_t)src;
        uint32_t la = (uint32_t)(uintptr_t)buf0;   // low 32 bits = LDS byte offset

        u32x4 g0;
        g0.x = 1u;                                  // count=1 (valid user D#)
        g0.y = la;                                  // lds_addr
        g0.z = (uint32_t)ga;                        // global_addr[31:0]
        g0.w = ((uint32_t)(ga >> 32) & 0x01FFFFFFu) // global_addr[56:32]
               | 0x80000000u;                       // type=2 ("image")

        i32x8 g1;
        g1[0] = (int)(2u << 16);                    // data_size = 4 bytes
        g1[1] = (int)((L0 & 0xFFFF) << 16);         // tensor_dim0[15:0]
        g1[2] = (int)(((L0 >> 16) & 0xFFFF) | (1 << 16)); // tensor_dim0[31:16], tensor_dim1=1
        g1[3] = (int)((L0 & 0xFFFF) << 16);         // tile_dim0 = 8192
        g1[4] = 0;                                  // tile_dim1=0 (1D), tile_dim2=0
        g1[5] = L0;                                 // tensor_dim0_stride
        g1[6] = 0;
        g1[7] = 0;

        i32x4 gz = {0, 0, 0, 0};
#if __clang_major__ >= 23
        i32x8 gz8 = {0, 0, 0, 0, 0, 0, 0, 0};
        __builtin_amdgcn_tensor_load_to_lds(g0, g1, gz, gz, gz8, 0);
#else
        __builtin_amdgcn_tensor_load_to_lds(g0, g1, gz, gz, 0);
#endif
        __builtin_amdgcn_s_wait_tensorcnt(0);
    }
#else
    for (int i = tid * 4; i < L0; i += NT * 4) {
        *(float4*)(buf0 + i) = *(const float4*)(src + i);
    }
#endif
    __syncthreads();

    // filter taps (tiny, L2-resident; scalar-loaded)
    const float a0 = w0[0], a1 = w0[1], a2 = w0[2];
    const float b0 = w1[0], b1 = w1[1], b2 = w1[2], b3 = w1[3], b4 = w1[4];
    const float c0 = w2[0], c1 = w2[1], c2 = w2[2], c3 = w2[3],
                c4 = w2[4], c5 = w2[5], c6 = w2[6];

    // ---- Phase A: scale-0 conv (k=3) -> out row 0 (NT b128 stores);
    //      s1 = avgpool2(buf0) ----
    for (int i = tid * 4; i < L0; i += NT * 4) {
        f32x4 v;
        #pragma unroll
        for (int e = 0; e < 4; ++e) {
            int idx = i + e;
            float l = (idx > 0)      ? buf0[idx - 1] : 0.0f;
            float m = buf0[idx];
            float q = (idx < L0 - 1) ? buf0[idx + 1] : 0.0f;
            v[e] = a0 * l + a1 * m + a2 * q;
        }
        __builtin_nontemporal_store(v, (f32x4*)(dst + i));
    }
    for (int j = tid; j < 4096; j += NT)
        s1[j] = 0.5f * (buf0[2 * j] + buf0[2 * j + 1]);
    __syncthreads();   // buf0 free now

    // ---- Phase B: f1 = conv5(s1) (4096) ----
    for (int j = tid; j < 4096; j += NT) {
        float acc = b2 * s1[j];
        acc += b0 * zget(s1, 4096, j - 2);
        acc += b1 * zget(s1, 4096, j - 1);
        acc += b3 * zget(s1, 4096, j + 1);
        acc += b4 * zget(s1, 4096, j + 2);
        f1[j] = acc;
    }
    __syncthreads();

    // ---- Phase C: p1 = avgpool2(f1) (2048); s2 = avgpool2(s1) (2048) ----
    for (int j = tid; j < 2048; j += NT) {
        p1[j] = 0.5f * (f1[2 * j] + f1[2 * j + 1]);
        s2[j] = 0.5f * (s1[2 * j] + s1[2 * j + 1]);
    }
    __syncthreads();   // f1 region & s1 free now

    // ---- Phase D: out row 1 = up2(up2(p1)) (NT b128 stores); f2 = conv7(s2) ----
    for (int i = tid * 4; i < L0; i += NT * 4) {
        f32x4 v;
        #pragma unroll
        for (int e = 0; e < 4; ++e) {
            int idx = i + e;
            // outer 2x upsample over virtual u1 (len 4096), u1 evaluated from p1
            float q = 0.5f * (float)idx - 0.25f;
            q = q < 0.0f ? 0.0f : q;
            int lo = (int)q;
            int hi = lo + 1; if (hi > 4095) hi = 4095;
            float w  = q - (float)lo;
            float va = up2_eval(p1, 2048, lo);
            float vb = up2_eval(p1, 2048, hi);
            v[e] = va + w * (vb - va);
        }
        __builtin_nontemporal_store(v, (f32x4*)(dst + L0 + i));
    }
    for (int j = tid; j < 2048; j += NT) {
        float acc = c3 * s2[j];
        acc += c0 * zget(s2, 2048, j - 3);
        acc += c1 * zget(s2, 2048, j - 2);
        acc += c2 * zget(s2, 2048, j - 1);
        acc += c4 * zget(s2, 2048, j + 1);
        acc += c5 * zget(s2, 2048, j + 2);
        acc += c6 * zget(s2, 2048, j + 3);
        f2[j] = acc;
    }
    __syncthreads();

    // ---- Phase E: p2 = avgpool2(f2) (1024) ----
    for (int j = tid; j < 1024; j += NT)
        p2[j] = 0.5f * (f2[2 * j] + f2[2 * j + 1]);
    __syncthreads();

    // ---- Phase F: out row 2 = interp4(up2(p2)) (NT b128 stores) ----
    for (int i = tid * 4; i < L0; i += NT * 4) {
        f32x4 v;
        #pragma unroll
        for (int e = 0; e < 4; ++e) {
            int idx = i + e;
            // 4x upsample over virtual c2arr (len 2048), evaluated from p2
            float q = 0.25f * (float)idx - 0.375f;
            q = q < 0.0f ? 0.0f : q;
            int lo = (int)q;
            int hi = lo + 1; if (hi > 2047) hi = 2047;
            float w  = q - (float)lo;
            float va = up2_eval(p2, 1024, lo);
            float vb = up2_eval(p2, 1024, hi);
            v[e] = va + w * (vb - va);
        }
        __builtin_nontemporal_store(v, (f32x4*)(dst + 2 * L0 + i));
    }
}

extern "C" void kernel_launch(void* const* d_in, const int* in_sizes, int n_in,
                              void* d_out, int out_size, void* d_ws, size_t ws_size,
                              hipStream_t stream) {
    (void)n_in; (void)out_size; (void)d_ws; (void)ws_size;
    const float* x  = (const float*)d_in[0];
    const float* w0 = (const float*)d_in[1];
    const float* w1 = (const float*)d_in[2];
    const float* w2 = (const float*)d_in[3];
    float* out = (float*)d_out;
    const int rows = in_sizes[0] / L0;   // 32*64 = 2048
    wavelet_fused_kernel<<<rows, NT, 0, stream>>>(x, w0, w1, w2, out);
}